// BetterGCN_42219528520184
// MI455X (gfx1250) — compile-verified
//
#include <hip/hip_runtime.h>
#include <cstdint>
#include <cstddef>

#define NN   100000
#define EE   3200000
#define FIN  512
#define FHID 16
#define FOUT 7

typedef __attribute__((ext_vector_type(2))) float v2f;
typedef __attribute__((ext_vector_type(8))) float v8f;

// ---------------------------------------------------------------------------
// 0) Init workspace: deg = 1.0 (self loop), agg1 = 0, agg2 = 0
// ---------------------------------------------------------------------------
__global__ void gcn_init(float* __restrict__ deg,
                         float* __restrict__ agg1,
                         float* __restrict__ agg2) {
    int i = blockIdx.x * 256 + threadIdx.x;
    if (i < NN)        deg[i]  = 1.0f;
    if (i < NN * FHID) agg1[i] = 0.0f;
    if (i < NN * FOUT) agg2[i] = 0.0f;
}

// ---------------------------------------------------------------------------
// 1) Degree: deg[col[e]] += 1   (self-loop already counted via init=1.0)
// ---------------------------------------------------------------------------
__global__ void gcn_degree(const int* __restrict__ col, float* __restrict__ deg) {
    int e = blockIdx.x * 256 + threadIdx.x;
    if (e < EE) atomicAdd(&deg[col[e]], 1.0f);
}

// ---------------------------------------------------------------------------
// 2) dinv = rsqrt(deg)  (deg >= 1 always, matches where(deg>0,...) )
// ---------------------------------------------------------------------------
__global__ void gcn_dinv(float* __restrict__ deg) {
    int i = blockIdx.x * 256 + threadIdx.x;
    if (i < NN) deg[i] = rsqrtf(deg[i]);
}

// ---------------------------------------------------------------------------
// 3) WMMA GEMM: h1 = x @ W1    (100000x512 @ 512x16, fp32)
//    One wave per 16-row tile; 8 waves (256 threads) per block.
//    W1 (32 KB) staged into LDS via GLOBAL_LOAD_ASYNC_TO_LDS_B128 (ASYNCcnt),
//    linear layout; B fragments are ds_load_2addr_b32 pairs (64 B apart).
//
//    A 16x4 f32 fragment (2 VGPRs): lane l<16 holds A[l][k],A[l][k+1];
//    lane l>=16 holds A[l-16][k+2],A[l-16][k+3].
//    B 4x16 f32 fragment mirrored along K in the lane halves.
//    C/D 16x16 f32: column = lane&15, rows = v + (lane>>4)*8.
// ---------------------------------------------------------------------------
__global__ void __launch_bounds__(256)
gcn_gemm1_wmma(const float* __restrict__ x,
               const float* __restrict__ W1,
               float* __restrict__ h1) {
    __shared__ float lds_w[FIN * FHID];   // 32 KB, linear (k*16 + c)

    // Async copy W1 -> LDS: 256 lanes x 8 iters x 16 B = 32768 B.
    {
        const uint64_t w1base = (uint64_t)(uintptr_t)W1;
        uint32_t ldsbase = (uint32_t)(uintptr_t)(&lds_w[0]);  // LDS offset bits
#pragma unroll
        for (int i = 0; i < 8; ++i) {
            uint32_t byteOff = (uint32_t)(i * 256 + threadIdx.x) * 16u;
            uint32_t ldsaddr = ldsbase + byteOff;
            asm volatile("global_load_async_to_lds_b128 %0, %1, %2"
                         :
                         : "v"(ldsaddr), "v"(byteOff), "s"(w1base)
                         : "memory");
        }
        asm volatile("s_wait_asynccnt 0x0" ::: "memory");
    }
    __syncthreads();

    const int wave = threadIdx.x >> 5;
    const int lane = threadIdx.x & 31;
    const int half = lane >> 4;     // 0: K pair low, 1: K pair high
    const int l16  = lane & 15;

    const int tile = blockIdx.x * 8 + wave;     // 16 rows per tile
    if (tile * 16 >= NN) return;                // whole wave uniform

    const float* arow = x + (size_t)(tile * 16 + l16) * FIN;

    v8f c = {};
#pragma unroll 4
    for (int k = 0; k < FIN; k += 4) {
        const int kb = k + half * 2;
        v2f a = *(const v2f*)(arow + kb);        // global b64
        v2f b;                                   // ds 2addr b32 (delta 16 dwords)
        b.x = lds_w[kb * FHID + l16];
        b.y = lds_w[kb * FHID + l16 + FHID];
        c = __builtin_amdgcn_wmma_f32_16x16x4_f32(
                /*neg_a=*/false, a, /*neg_b=*/false, b,
                /*c_mod=*/(short)0, c, /*reuse_a=*/false, /*reuse_b=*/false);
    }

    float* hrow = h1 + ((size_t)tile * 16 + half * 8) * FHID + l16;
#pragma unroll
    for (int v = 0; v < 8; ++v) hrow[(size_t)v * FHID] = c[v];
}

// ---------------------------------------------------------------------------
// 4) Edge aggregation layer 1: agg1[c][:] += dinv[r]*dinv[c] * h1[r][:]
//    h1/agg1 are 6.4 MB each -> resident in 192 MB L2, atomics stay on-chip.
// ---------------------------------------------------------------------------
__global__ void gcn_agg1(const int* __restrict__ row, const int* __restrict__ col,
                         const float* __restrict__ dinv,
                         const float* __restrict__ h1,
                         float* __restrict__ agg1) {
    int e = blockIdx.x * 256 + threadIdx.x;
    if (e >= EE) return;
    int r = row[e], cNode = col[e];
    float w = dinv[r] * dinv[cNode];
    const float4* hr = (const float4*)(h1 + (size_t)r * FHID);
    float* out = agg1 + (size_t)cNode * FHID;
#pragma unroll
    for (int q = 0; q < 4; ++q) {
        float4 v = hr[q];
        atomicAdd(out + 4 * q + 0, w * v.x);
        atomicAdd(out + 4 * q + 1, w * v.y);
        atomicAdd(out + 4 * q + 2, w * v.z);
        atomicAdd(out + 4 * q + 3, w * v.w);
    }
}

// ---------------------------------------------------------------------------
// 5) Per-node epilogue1 + layer-2 transform:
//    o = relu(agg1[i] + dinv[i]^2 * h1[i] + b1);  h2[i] = o @ W2
// ---------------------------------------------------------------------------
__global__ void gcn_post1(const float* __restrict__ dinv,
                          const float* __restrict__ h1,
                          const float* __restrict__ agg1,
                          const float* __restrict__ b1,
                          const float* __restrict__ W2,
                          float* __restrict__ h2) {
    __shared__ float sW2[FHID * FOUT];
    __shared__ float sb1[FHID];
    if (threadIdx.x < FHID * FOUT) sW2[threadIdx.x] = W2[threadIdx.x];
    if (threadIdx.x < FHID)        sb1[threadIdx.x] = b1[threadIdx.x];
    __syncthreads();

    int i = blockIdx.x * 256 + threadIdx.x;
    if (i >= NN) return;
    float di = dinv[i];
    float selfw = di * di;

    float o[FHID];
#pragma unroll
    for (int f = 0; f < FHID; ++f) {
        float t = agg1[(size_t)i * FHID + f]
                + selfw * h1[(size_t)i * FHID + f] + sb1[f];
        o[f] = fmaxf(t, 0.0f);
    }
#pragma unroll
    for (int j = 0; j < FOUT; ++j) {
        float s = 0.0f;
#pragma unroll
        for (int f = 0; f < FHID; ++f) s = fmaf(o[f], sW2[f * FOUT + j], s);
        h2[(size_t)i * FOUT + j] = s;
    }
}

// ---------------------------------------------------------------------------
// 6) Edge aggregation layer 2: agg2[c][:] += w * h2[r][:]   (7 feats)
// ---------------------------------------------------------------------------
__global__ void gcn_agg2(const int* __restrict__ row, const int* __restrict__ col,
                         const float* __restrict__ dinv,
                         const float* __restrict__ h2,
                         float* __restrict__ agg2) {
    int e = blockIdx.x * 256 + threadIdx.x;
    if (e >= EE) return;
    int r = row[e], cNode = col[e];
    float w = dinv[r] * dinv[cNode];
    const float* hr = h2 + (size_t)r * FOUT;
    float* out = agg2 + (size_t)cNode * FOUT;
#pragma unroll
    for (int j = 0; j < FOUT; ++j) atomicAdd(out + j, w * hr[j]);
}

// ---------------------------------------------------------------------------
// 7) Final: v = agg2[i] + dinv^2*h2[i] + b2;  out = log_softmax(v)
// ---------------------------------------------------------------------------
__global__ void gcn_final(const float* __restrict__ dinv,
                          const float* __restrict__ h2,
                          const float* __restrict__ agg2,
                          const float* __restrict__ b2,
                          float* __restrict__ out) {
    int i = blockIdx.x * 256 + threadIdx.x;
    if (i >= NN) return;
    float di = dinv[i];
    float selfw = di * di;

    float v[FOUT];
    float m = -3.402823466e38f;
#pragma unroll
    for (int j = 0; j < FOUT; ++j) {
        v[j] = agg2[(size_t)i * FOUT + j]
             + selfw * h2[(size_t)i * FOUT + j] + b2[j];
        m = fmaxf(m, v[j]);
    }
    float s = 0.0f;
#pragma unroll
    for (int j = 0; j < FOUT; ++j) s += __expf(v[j] - m);
    float lse = __logf(s);
#pragma unroll
    for (int j = 0; j < FOUT; ++j)
        out[(size_t)i * FOUT + j] = v[j] - m - lse;
}

// ---------------------------------------------------------------------------
// Launch
// ---------------------------------------------------------------------------
extern "C" void kernel_launch(void* const* d_in, const int* in_sizes, int n_in,
                              void* d_out, int out_size, void* d_ws, size_t ws_size,
                              hipStream_t stream) {
    (void)in_sizes; (void)n_in; (void)out_size; (void)ws_size;

    const float* x    = (const float*)d_in[0];
    const int*   eidx = (const int*)  d_in[1];     // [2, E]
    const float* W1   = (const float*)d_in[2];
    const float* b1   = (const float*)d_in[3];
    const float* W2   = (const float*)d_in[4];
    const float* b2   = (const float*)d_in[5];
    float*       out  = (float*)d_out;

    const int* row = eidx;        // sources
    const int* col = eidx + EE;   // targets

    // Workspace layout (floats): deg[N] | h1[N*16] | agg1[N*16] | h2[N*7] | agg2[N*7]
    float* ws   = (float*)d_ws;
    float* deg  = ws;                       // doubles as dinv after step 2
    float* h1   = ws + (size_t)NN;
    float* agg1 = ws + (size_t)NN * 17;
    float* h2   = ws + (size_t)NN * 33;
    float* agg2 = ws + (size_t)NN * 40;

    const int edgeBlocks = (EE + 255) / 256;          // 12500
    const int nodeBlocks = (NN + 255) / 256;          // 391
    const int initBlocks = (NN * FHID + 255) / 256;   // 6250
    const int gemmBlocks = (NN / 16 + 7) / 8;         // 6250 tiles / 8 waves -> 782

    gcn_init   <<<initBlocks, 256, 0, stream>>>(deg, agg1, agg2);
    gcn_degree <<<edgeBlocks, 256, 0, stream>>>(col, deg);
    gcn_dinv   <<<nodeBlocks, 256, 0, stream>>>(deg);
    gcn_gemm1_wmma<<<gemmBlocks, 256, 0, stream>>>(x, W1, h1);
    gcn_agg1   <<<edgeBlocks, 256, 0, stream>>>(row, col, deg, h1, agg1);
    gcn_post1  <<<nodeBlocks, 256, 0, stream>>>(deg, h1, agg1, b1, W2, h2);
    gcn_agg2   <<<edgeBlocks, 256, 0, stream>>>(row, col, deg, h2, agg2);
    gcn_final  <<<nodeBlocks, 256, 0, stream>>>(deg, h2, agg2, b2, out);
}